// GATLayer_352187318572
// MI455X (gfx1250) — compile-verified
//
#include <hip/hip_runtime.h>
#include <hip/hip_bf16.h>

typedef __attribute__((ext_vector_type(2))) float v2f;
typedef __attribute__((ext_vector_type(8))) float v8f;

#define DIN  128
#define EDIM 64
#define DK   (DIN + EDIM)   // 192
#define DOUT 128

// ---------------------------------------------------------------------------
// Kernel 1: per-node attention pre-projection.
//   a_src[n] = nfeats[n] . attn_w[0:128]
//   a_dst[n] = nfeats[n] . attn_w[128:256]
// One wave32 per node; each lane owns a float4 (128/32 = 4 elems), then a
// 5-step shfl_xor butterfly reduction (wave32!).
// ---------------------------------------------------------------------------
__global__ __launch_bounds__(256) void node_precompute_kernel(
    const float* __restrict__ nfeats, const float* __restrict__ attn_w,
    float* __restrict__ a_src, float* __restrict__ a_dst, int n_nodes) {
  int node = (int)((blockIdx.x * blockDim.x + threadIdx.x) >> 5);
  int lane = threadIdx.x & 31;
  if (node >= n_nodes) return;
  const float4 x  = ((const float4*)(nfeats + (size_t)node * DIN))[lane];
  const float4 ws = ((const float4*)attn_w)[lane];
  const float4 wd = ((const float4*)(attn_w + DIN))[lane];
  float ps = x.x * ws.x + x.y * ws.y + x.z * ws.z + x.w * ws.w;
  float pd = x.x * wd.x + x.y * wd.y + x.z * wd.z + x.w * wd.w;
#pragma unroll
  for (int off = 16; off > 0; off >>= 1) {
    ps += __shfl_xor(ps, off, 32);
    pd += __shfl_xor(pd, off, 32);
  }
  if (lane == 0) { a_src[node] = ps; a_dst[node] = pd; }
}

// ---------------------------------------------------------------------------
// Kernel 2: edge score + segment max.
//   e = relu(a_src[src] + a_dst[dst] + b)  (>= 0)
// emax initialized to 0.0f; since e >= 0, uint-bit-pattern atomicMax is an
// exact float max, and isolated nodes keep 0 (matches the isneginf->0 fixup).
// ---------------------------------------------------------------------------
__global__ void edge_score_kernel(const int* __restrict__ src,
                                  const int* __restrict__ dst,
                                  const float* __restrict__ a_src,
                                  const float* __restrict__ a_dst,
                                  const float* __restrict__ attn_b,
                                  float* __restrict__ ebuf,
                                  float* __restrict__ emax, int n_edges) {
  int i = (int)(blockIdx.x * blockDim.x + threadIdx.x);
  if (i >= n_edges) return;
  float e = fmaxf(a_src[src[i]] + a_dst[dst[i]] + attn_b[0], 0.0f);
  ebuf[i] = e;
  atomicMax((unsigned int*)(emax + dst[i]), __float_as_uint(e));
}

// ---------------------------------------------------------------------------
// Kernel 3: ex = exp(e - emax[dst]); denom[dst] += ex  (ex overwrites ebuf)
// ---------------------------------------------------------------------------
__global__ void edge_exp_kernel(const int* __restrict__ dst,
                                const float* __restrict__ emax,
                                float* __restrict__ ebuf,
                                float* __restrict__ denom, int n_edges) {
  int i = (int)(blockIdx.x * blockDim.x + threadIdx.x);
  if (i >= n_edges) return;
  int d = dst[i];
  float ex = __expf(ebuf[i] - emax[d]);
  ebuf[i] = ex;
  atomicAdd(&denom[d], ex);
}

// ---------------------------------------------------------------------------
// Kernel 4: z[dst] += alpha * efeats   (the 205 MB streaming pass)
// 64 threads per edge (one lane per feature), 4 edges per 256-thread block.
// efeats read is perfectly coalesced; z atomics land in L2 (z = 12.8 MB).
// ---------------------------------------------------------------------------
__global__ __launch_bounds__(256) void edge_aggregate_kernel(
    const float* __restrict__ efeats, const int* __restrict__ dst,
    const float* __restrict__ ebuf, const float* __restrict__ denom,
    float* __restrict__ z, int n_edges) {
  int edge = (int)(blockIdx.x * 4 + (threadIdx.x >> 6));
  int t = threadIdx.x & 63;
  if (edge >= n_edges) return;
  if (edge + 8 < n_edges)  // stream-ahead hint -> global_prefetch_b8
    __builtin_prefetch(efeats + (size_t)(edge + 8) * EDIM + t, 0, 1);
  int d = dst[edge];
  float den = denom[d];
  float alpha = ebuf[edge] / (den > 0.0f ? den : 1.0f);
  atomicAdd(&z[(size_t)d * EDIM + t],
            alpha * efeats[(size_t)edge * EDIM + t]);
}

// ---------------------------------------------------------------------------
// Kernel 5: out = relu([nfeats || z] @ W^T + b) via V_WMMA_F32_16X16X4_F32.
// Block = 256 threads = 8 wave32s; block owns a 16-node M-tile, wave w owns
// output columns [16w, 16w+16). K = 192 -> 48 WMMAs (full f32 precision).
//
// A (16x4 f32) lane layout:  M = lane&15 ; lanes 0-15 hold K{0,1},
//                            lanes 16-31 hold K{2,3}      (2 VGPRs)
// B (4x16 f32) symmetric:    N = lane&15 ; same K split   (2 VGPRs)
// C/D (16x16 f32, 8 VGPRs):  VGPR v -> row v + 8*(lane>>4), col = lane&15
// EXEC must be all-ones: grid covers only full 16-node tiles (50000 % 16 == 0;
// a scalar tail kernel handles any remainder in the general case).
// ---------------------------------------------------------------------------
__global__ __launch_bounds__(256) void apply_wmma_kernel(
    const float* __restrict__ nfeats, const float* __restrict__ z,
    const float* __restrict__ W, const float* __restrict__ bias,
    float* __restrict__ out, int tiles) {
  const int tile = (int)blockIdx.x;
  const int wave = threadIdx.x >> 5;
  const int lane = threadIdx.x & 31;
  const int r    = lane & 15;           // A row / B col within tile
  const int kg   = (lane >> 4) << 1;    // K sub-offset: 0 or 2
  const size_t node = (size_t)tile * 16 + r;
  const int col  = (wave << 4) + r;     // global output column

  const float* __restrict__ Arow = nfeats + node * DIN;
  const float* __restrict__ Zrow = z + node * EDIM;
  const float* __restrict__ Wrow = W + (size_t)col * DK;  // B[k][n] = W[n][k]

  v8f acc = {};
#pragma unroll
  for (int kb = 0; kb < DIN; kb += 4) {   // K = 0..127 from nfeats
    v2f a, b;
    a.x = Arow[kb + kg];      a.y = Arow[kb + kg + 1];
    b.x = Wrow[kb + kg];      b.y = Wrow[kb + kg + 1];
    acc = __builtin_amdgcn_wmma_f32_16x16x4_f32(
        false, a, false, b, (short)0, acc, false, false);
  }
#pragma unroll
  for (int kb = 0; kb < EDIM; kb += 4) {  // K = 128..191 from z
    v2f a, b;
    a.x = Zrow[kb + kg];          a.y = Zrow[kb + kg + 1];
    b.x = Wrow[DIN + kb + kg];    b.y = Wrow[DIN + kb + kg + 1];
    acc = __builtin_amdgcn_wmma_f32_16x16x4_f32(
        false, a, false, b, (short)0, acc, false, false);
  }

  const float bv = bias[col];
  const int rbase = (lane >> 4) << 3;     // 0 or 8
  float* __restrict__ outTile = out + (size_t)tile * 16 * DOUT;
#pragma unroll
  for (int v = 0; v < 8; ++v) {
    outTile[(size_t)(rbase + v) * DOUT + col] = fmaxf(acc[v] + bv, 0.0f);
  }
  (void)tiles;
}

// Scalar tail for n_nodes % 16 != 0 (never launches for N = 50000).
__global__ void apply_tail_kernel(const float* __restrict__ nfeats,
                                  const float* __restrict__ z,
                                  const float* __restrict__ W,
                                  const float* __restrict__ bias,
                                  float* __restrict__ out,
                                  int node0, int n_nodes) {
  int idx = (int)(blockIdx.x * blockDim.x + threadIdx.x);
  int node = node0 + (idx >> 7);
  int c = idx & (DOUT - 1);
  if (node >= n_nodes) return;
  const float* Wr = W + (size_t)c * DK;
  float s = bias[c];
  for (int k = 0; k < DIN; ++k)  s += nfeats[(size_t)node * DIN + k] * Wr[k];
  for (int k = 0; k < EDIM; ++k) s += z[(size_t)node * EDIM + k] * Wr[DIN + k];
  out[(size_t)node * DOUT + c] = fmaxf(s, 0.0f);
}

// ---------------------------------------------------------------------------
extern "C" void kernel_launch(void* const* d_in, const int* in_sizes, int n_in,
                              void* d_out, int out_size, void* d_ws,
                              size_t ws_size, hipStream_t stream) {
  const float* nfeats = (const float*)d_in[0];
  const float* efeats = (const float*)d_in[1];
  const int*   src    = (const int*)d_in[2];
  const int*   dst    = (const int*)d_in[3];
  const float* W      = (const float*)d_in[4];
  const float* bias   = (const float*)d_in[5];
  const float* attn_w = (const float*)d_in[6];
  const float* attn_b = (const float*)d_in[7];
  float* out = (float*)d_out;

  const int n_nodes = in_sizes[0] / DIN;   // 50000
  const int n_edges = in_sizes[2];         // 800000

  // Workspace layout (floats): a_src[N] a_dst[N] ebuf[E] | emax[N] denom[N] z[N*64]
  float* ws    = (float*)d_ws;
  float* a_src = ws;
  float* a_dst = a_src + n_nodes;
  float* ebuf  = a_dst + n_nodes;
  float* emax  = ebuf + n_edges;
  float* denom = emax + n_nodes;
  float* zbuf  = denom + n_nodes;

  // Zero the accumulator region (emax, denom, z are contiguous).
  hipMemsetAsync(emax, 0, (size_t)n_nodes * (2 + EDIM) * sizeof(float), stream);

  {  // per-node attention projections: 8 wave32 per 256-thread block
    int blocks = (n_nodes + 7) / 8;
    node_precompute_kernel<<<blocks, 256, 0, stream>>>(nfeats, attn_w, a_src,
                                                       a_dst, n_nodes);
  }
  {
    int blocks = (n_edges + 255) / 256;
    edge_score_kernel<<<blocks, 256, 0, stream>>>(src, dst, a_src, a_dst,
                                                  attn_b, ebuf, emax, n_edges);
    edge_exp_kernel<<<blocks, 256, 0, stream>>>(dst, emax, ebuf, denom,
                                                n_edges);
  }
  {
    int blocks = (n_edges + 3) / 4;
    edge_aggregate_kernel<<<blocks, 256, 0, stream>>>(efeats, dst, ebuf, denom,
                                                      zbuf, n_edges);
  }
  int tiles = n_nodes / 16;
  if (tiles > 0)
    apply_wmma_kernel<<<tiles, 256, 0, stream>>>(nfeats, zbuf, W, bias, out,
                                                 tiles);
  int rem = n_nodes & 15;
  if (rem > 0) {
    int threads = rem * DOUT;
    apply_tail_kernel<<<(threads + 255) / 256, 256, 0, stream>>>(
        nfeats, zbuf, W, bias, out, tiles * 16, n_nodes);
  }
}